// UnitaryR3Ansatz_18846316495450
// MI455X (gfx1250) — compile-verified
//
#include <hip/hip_runtime.h>

typedef __attribute__((ext_vector_type(2))) float v2f;
typedef __attribute__((ext_vector_type(8))) float v8f;

#define NQ      10
#define DIM     1024
#define LAYERS  8
#define COPIES  32
#define GSTRIDE 528   // 256 (G0) + 256 (G1) + 16 (G2) complex entries per (copy,layer)

__device__ __forceinline__ float2 cmul(float2 a, float2 b) {
    return make_float2(a.x * b.x - a.y * b.y, a.x * b.y + a.y * b.x);
}
__device__ __forceinline__ float2 cfma(float2 a, float2 b, float2 c) {
    return make_float2(fmaf(a.x, b.x, fmaf(-a.y, b.y, c.x)),
                       fmaf(a.x, b.y, fmaf( a.y, b.x, c.y)));
}

// ---------------------------------------------------------------------------
// Kernel 1: build per-(copy,layer) grouped gate matrices
//   G0 = kron(r0,r1,r2,r3)  (16x16)   -> acts on index bits [9:6]
//   G1 = kron(r4,r5,r6,r7)  (16x16)   -> acts on index bits [5:2]
//   G2 = kron(r8,r9)        (4x4)     -> acts on index bits [1:0]
// ---------------------------------------------------------------------------
__global__ __launch_bounds__(256)
void build_gates_kernel(const float* __restrict__ params, float2* __restrict__ G)
{
    __shared__ float2 r[NQ][2][2];
    const int c = blockIdx.x, l = blockIdx.y, t = threadIdx.x;

    if (t < NQ) {
        const float* p = params + ((c * LAYERS + l) * NQ + t) * 3;
        const float omega = p[0], theta = p[1], phi = p[2];
        float sh, ch, s0, c0, s1, c1;
        __sincosf(0.5f * theta, &sh, &ch);
        __sincosf(-0.5f * (phi + omega), &s0, &c0);   // e^{-i(phi+omega)/2} = c0 + i s0
        __sincosf( 0.5f * (phi - omega), &s1, &c1);   // e^{ i(phi-omega)/2} = c1 + i s1
        r[t][0][0] = make_float2( c0 * ch,  s0 * ch);   // m11
        r[t][0][1] = make_float2(-c1 * sh, -s1 * sh);   // m12
        r[t][1][0] = make_float2( c1 * sh, -s1 * sh);   // m21
        r[t][1][1] = make_float2( c0 * ch, -s0 * ch);   // m22
    }
    __syncthreads();

    float2* out = G + (c * LAYERS + l) * GSTRIDE;
    const int m = t >> 4, n = t & 15;

    float2 v0 = r[0][(m >> 3) & 1][(n >> 3) & 1];
    v0 = cmul(v0, r[1][(m >> 2) & 1][(n >> 2) & 1]);
    v0 = cmul(v0, r[2][(m >> 1) & 1][(n >> 1) & 1]);
    v0 = cmul(v0, r[3][m & 1][n & 1]);
    out[t] = v0;

    float2 v1 = r[4][(m >> 3) & 1][(n >> 3) & 1];
    v1 = cmul(v1, r[5][(m >> 2) & 1][(n >> 2) & 1]);
    v1 = cmul(v1, r[6][(m >> 1) & 1][(n >> 1) & 1]);
    v1 = cmul(v1, r[7][m & 1][n & 1]);
    out[256 + t] = v1;

    if (t < 16) {
        const int m2 = t >> 2, n2 = t & 3;
        out[512 + t] = cmul(r[8][(m2 >> 1) & 1][(n2 >> 1) & 1], r[9][m2 & 1][n2 & 1]);
    }
}

// ---------------------------------------------------------------------------
// Apply a 16x16 complex gate via V_WMMA_F32_16X16X4_F32.
// State in LDS: s[v*1024 + i], 2 vectors per block -> 128 columns = 8 tiles,
// one 16-column tile per wave. Columns are disjoint across waves.
//   stage 0: gate on bits [9:6], X[k][n]: n->(v=n>>6, b=n&63), addr = v*1024 + k*64 + b
//   stage 1: gate on bits [5:2], n->(v, a=(n>>2)&15, t=n&3), addr = v*1024 + a*64 + k*4 + t
// Complex GEMM: Yr = Gr·Xr + (−Gi)·Xi ; Yi = Gr·Xi + Gi·Xr  (2 f32 accumulators)
// ---------------------------------------------------------------------------
__device__ __forceinline__ void apply16(const float2* __restrict__ g,
                                        float2* __restrict__ s,
                                        int lane, int wave, int stage)
{
    const int hi = lane >> 4;          // K-half select (ISA A/B layout)
    const int nl = lane & 15;          // A row m / B column n
    const int n  = wave * 16 + nl;     // global column 0..127
    int cb, ks;
    if (stage == 0) { cb = (n >> 6) * DIM + (n & 63);                        ks = 64; }
    else            { cb = (n >> 6) * DIM + ((n >> 2) & 15) * 64 + (n & 3);  ks = 4;  }

    v8f accr = {};
    v8f acci = {};
#pragma unroll
    for (int kc = 0; kc < 4; ++kc) {
        const int k0 = 4 * kc + hi * 2;            // this lane's K pair within chunk
        const float2 ga = g[nl * 16 + k0];         // G[m][k0]
        const float2 gb = g[nl * 16 + k0 + 1];     // G[m][k0+1]
        v2f ar  = {  ga.x,  gb.x };
        v2f ai  = {  ga.y,  gb.y };
        v2f ain = { -ga.y, -gb.y };
        const float2 xa = s[cb + k0 * ks];         // X[k0][n]
        const float2 xb = s[cb + (k0 + 1) * ks];   // X[k0+1][n]
        v2f br = { xa.x, xb.x };
        v2f bi = { xa.y, xb.y };
        accr = __builtin_amdgcn_wmma_f32_16x16x4_f32(false, ar,  false, br, (short)0, accr, false, false);
        accr = __builtin_amdgcn_wmma_f32_16x16x4_f32(false, ain, false, bi, (short)0, accr, false, false);
        acci = __builtin_amdgcn_wmma_f32_16x16x4_f32(false, ar,  false, bi, (short)0, acci, false, false);
        acci = __builtin_amdgcn_wmma_f32_16x16x4_f32(false, ai,  false, br, (short)0, acci, false, false);
    }
#pragma unroll
    for (int d = 0; d < 8; ++d) {                  // D layout: row = d + 8*hi, col = nl
        const int m = d + hi * 8;
        s[cb + m * ks] = make_float2(accr[d], acci[d]);
    }
}

// ---------------------------------------------------------------------------
// Kernel 2: state-vector simulation. 64 blocks = 32 copies x (2 vectors/block).
// Gate blocks are double-buffered: layer l+1's gates are loaded into registers
// at the top of iteration l (global loads issued early), and stored into the
// alternate LDS buffer only after all WMMA/DS compute — the s_wait_loadcnt
// lands behind ~32 matrix ops per wave, hiding global latency entirely.
// ---------------------------------------------------------------------------
__global__ __launch_bounds__(256)
void qsim_kernel(const float* __restrict__ input, const float2* __restrict__ Gg,
                 float2* __restrict__ out)
{
    __shared__ float2 s[2 * DIM];       // 16 KB: two complex state vectors
    __shared__ float2 g[2][GSTRIDE];    // 8.4 KB: double-buffered layer gates

    const int c    = blockIdx.x >> 1;
    const int vb0  = (blockIdx.x & 1) * 2;
    const int t    = threadIdx.x;
    const int lane = t & 31;
    const int wave = t >> 5;

    // initial state: complex(input, 0); preload layer-0 gates
    {
        const float2* src0 = Gg + (c * LAYERS) * GSTRIDE;
        g[0][t]       = src0[t];
        g[0][t + 256] = src0[t + 256];
        if (t < 16) g[0][t + 512] = src0[t + 512];
    }
    for (int idx = t; idx < 2 * DIM; idx += 256) {
        const int v = idx >> 10, i = idx & (DIM - 1);
        s[idx] = make_float2(input[(c * 4 + vb0 + v) * DIM + i], 0.0f);
    }
    __syncthreads();

    for (int l = 0; l < LAYERS; ++l) {
        const int cur = l & 1;

        // early-issue next layer's gate loads into registers (latency hidden
        // behind the WMMA stages below)
        const bool have_next = (l + 1) < LAYERS;
        float2 n0 = make_float2(0.f, 0.f), n1 = n0, n2 = n0;
        if (have_next) {
            const float2* srcn = Gg + (c * LAYERS + l + 1) * GSTRIDE;
            n0 = srcn[t];
            n1 = srcn[t + 256];
            if (t < 16) n2 = srcn[t + 512];
        }

        apply16(g[cur],       s, lane, wave, 0);   // qubits 0-3 (WMMA)
        __syncthreads();
        apply16(g[cur] + 256, s, lane, wave, 1);   // qubits 4-7 (WMMA)
        __syncthreads();

        // qubits 8-9: 4x4 complex gate on fastest 2 bits (one row per thread slot)
        {
            const float2* g2 = g[cur] + 512;
#pragma unroll
            for (int rr = 0; rr < 2; ++rr) {
                const int R = t + rr * 256;                    // 0..511 rows
                const int base = (R >> 8) * DIM + (R & 255) * 4;
                const float2 x0 = s[base + 0], x1 = s[base + 1];
                const float2 x2 = s[base + 2], x3 = s[base + 3];
#pragma unroll
                for (int m = 0; m < 4; ++m) {
                    float2 acc = cmul(g2[m * 4 + 0], x0);
                    acc = cfma(g2[m * 4 + 1], x1, acc);
                    acc = cfma(g2[m * 4 + 2], x2, acc);
                    acc = cfma(g2[m * 4 + 3], x3, acc);
                    s[base + m] = acc;
                }
            }
        }
        __syncthreads();

        // entangling block = CNOT chain permutation: y[i] = x[f8(...f0(i))]
        // qubit k <-> bit (9-k); f_k: if bit(9-k) set, flip bit(8-k), k=0..8
        float2 tmp[8];
#pragma unroll
        for (int ii = 0; ii < 8; ++ii) {
            const int idx = t + ii * 256;
            const int v = idx >> 10;
            unsigned j = (unsigned)(idx & (DIM - 1));
#pragma unroll
            for (int p = 9; p >= 1; --p)
                if ((j >> p) & 1u) j ^= (1u << (p - 1));
            tmp[ii] = s[v * DIM + (int)j];
        }
        __syncthreads();
#pragma unroll
        for (int ii = 0; ii < 8; ++ii) s[t + ii * 256] = tmp[ii];

        // commit next layer's gates to the alternate LDS buffer (loads have
        // had the whole layer's compute to complete)
        if (have_next) {
            g[cur ^ 1][t]       = n0;
            g[cur ^ 1][t + 256] = n1;
            if (t < 16) g[cur ^ 1][t + 512] = n2;
        }
        __syncthreads();
    }

    // write interleaved complex64 output
    for (int idx = t; idx < 2 * DIM; idx += 256) {
        const int v = idx >> 10, i = idx & (DIM - 1);
        out[(c * 4 + vb0 + v) * DIM + i] = s[idx];
    }
}

extern "C" void kernel_launch(void* const* d_in, const int* in_sizes, int n_in,
                              void* d_out, int out_size, void* d_ws, size_t ws_size,
                              hipStream_t stream)
{
    const float* input  = (const float*)d_in[0];   // 128 x 1024 f32
    const float* params = (const float*)d_in[1];   // 32 x 8 x 10 x 3 f32
    float2* G = (float2*)d_ws;                     // 32*8*528 complex = ~1.03 MB

    build_gates_kernel<<<dim3(COPIES, LAYERS), 256, 0, stream>>>(params, G);
    qsim_kernel<<<COPIES * 2, 256, 0, stream>>>(input, G, (float2*)d_out);
}